// Attention_74268574482673
// MI455X (gfx1250) — compile-verified
//
#include <hip/hip_runtime.h>
#include <hip/hip_bf16.h>

// ---------------------------------------------------------------------------
// Shapes from the reference: S=256, B=32, T=24, D=512
// ---------------------------------------------------------------------------
constexpr int kS = 256;
constexpr int kB = 32;
constexpr int kT = 24;
constexpr int kD = 512;

typedef float v2f __attribute__((ext_vector_type(2)));
typedef float v8f __attribute__((ext_vector_type(8)));

// Native fp32 WMMA: D(16x16,f32) += A(16x4,f32) * B(4x16,f32)
__device__ __forceinline__ v8f wmma4(v2f a, v2f b, v8f c) {
  return __builtin_amdgcn_wmma_f32_16x16x4_f32(
      /*neg_a=*/false, a, /*neg_b=*/false, b,
      /*c_mod=*/(short)0, c, /*reuse_a=*/false, /*reuse_b=*/false);
}

// ---------------------------------------------------------------------------
// Big GEMM: out[M=8192, N=512] = A0*W0^T (+ A1*W1^T) + b0 (+ b1)
// A matrices are contiguous [M, 512]; W are [512 out][512 in] row-major.
// Block = 256 threads (8 waves). Block owns rows [m0, m0+16), all 512 cols;
// wave w owns cols [64w, 64w+64) as four 16x16 WMMA accumulators.
// A stripe staged in LDS (16x512 f32 = 32 KB).
// remap: output row r = s*kB + b  is stored at row b*kS + s  ([S,B]->[B,S]).
// ---------------------------------------------------------------------------
__global__ __launch_bounds__(256) void big_gemm_kernel(
    const float* __restrict__ A0, const float* __restrict__ W0,
    const float* __restrict__ b0,
    const float* __restrict__ A1, const float* __restrict__ W1,
    const float* __restrict__ b1,
    float* __restrict__ out, int remap) {
  __shared__ float ldsA[16 * kD];

  const int tid  = threadIdx.x;
  const int lane = tid & 31;
  const int wave = tid >> 5;
  const int half = lane >> 4;   // 0: K pair {0,1}, 1: K pair {2,3}
  const int l16  = lane & 15;
  const int m0   = blockIdx.x * 16;

  v8f acc[4] = {};

  const int npasses = (A1 != nullptr) ? 2 : 1;
  for (int p = 0; p < npasses; ++p) {
    const float* __restrict__ A = p ? A1 : A0;
    const float* __restrict__ W = p ? W1 : W0;
    if (p) __syncthreads();  // all waves done reading LDS from pass 0
    // Cooperative stage of the 16x512 A stripe (2048 float4 loads).
#pragma unroll
    for (int it = 0; it < 8; ++it) {
      int f  = tid + it * 256;       // float4 index in [0, 2048)
      int rr = f >> 7;               // row 0..15
      int cc = (f & 127) << 2;       // col 0..508
      *(float4*)&ldsA[rr * kD + cc] =
          *(const float4*)&A[(size_t)(m0 + rr) * kD + cc];
    }
    __syncthreads();

    const int n_base = wave * 64;
    for (int k0 = 0; k0 < kD; k0 += 4) {
      const int kk = k0 + half * 2;
      v2f a = *(const v2f*)&ldsA[l16 * kD + kk];  // ds_load_b64
#pragma unroll
      for (int j = 0; j < 4; ++j) {
        const int n = n_base + j * 16 + l16;
        v2f bf = *(const v2f*)&W[(size_t)n * kD + kk];  // B[k][n] = W[n][k]
        acc[j] = wmma4(a, bf, acc[j]);
      }
    }
  }

  // Store with bias; C layout: vgpr v -> m = m0 + v + 8*half, col n = l16.
#pragma unroll
  for (int j = 0; j < 4; ++j) {
    const int n    = wave * 64 + j * 16 + l16;
    const float bv = b0[n] + (b1 ? b1[n] : 0.0f);
#pragma unroll
    for (int v = 0; v < 8; ++v) {
      const int m = m0 + v + half * 8;
      size_t off;
      if (remap) {
        const int s = m >> 5;       // m = s*kB + b, kB = 32
        const int b = m & 31;
        off = ((size_t)b * kS + s) * kD + n;
      } else {
        off = (size_t)m * kD + n;
      }
      out[off] = acc[j][v] + bv;
    }
  }
}

// ---------------------------------------------------------------------------
// Small GEMM: out[32, 512] = A0*W0^T (+ A1*W1^T) + b0 (+ b1), opt. tanh.
// One wave per 16x16 output tile; grid = (2, 32).
// ---------------------------------------------------------------------------
__global__ __launch_bounds__(32) void small_gemm_kernel(
    const float* __restrict__ A0, const float* __restrict__ W0,
    const float* __restrict__ b0,
    const float* __restrict__ A1, const float* __restrict__ W1,
    const float* __restrict__ b1,
    float* __restrict__ out, int apply_tanh) {
  const int lane = threadIdx.x & 31;
  const int half = lane >> 4;
  const int l16  = lane & 15;
  const int m0   = blockIdx.x * 16;
  const int n0   = blockIdx.y * 16;

  v8f acc = {};
  const int npasses = (A1 != nullptr) ? 2 : 1;
  for (int p = 0; p < npasses; ++p) {
    const float* __restrict__ A = p ? A1 : A0;
    const float* __restrict__ W = p ? W1 : W0;
    for (int k0 = 0; k0 < kD; k0 += 4) {
      const int kk = k0 + half * 2;
      v2f a  = *(const v2f*)&A[(size_t)(m0 + l16) * kD + kk];
      v2f bf = *(const v2f*)&W[(size_t)(n0 + l16) * kD + kk];
      acc = wmma4(a, bf, acc);
    }
  }
  const int n    = n0 + l16;
  const float bv = b0[n] + (b1 ? b1[n] : 0.0f);
#pragma unroll
  for (int v = 0; v < 8; ++v) {
    const int m = m0 + v + half * 8;
    float val   = acc[v] + bv;
    if (apply_tanh) val = tanhf(val);
    out[(size_t)m * kD + n] = val;
  }
}

// ---------------------------------------------------------------------------
// logits[b*S+s] = sum_e tanh(co_dm[b,s,e] + add[b,e]) * ms_w[e] + ms_b
// One wave per (b,s) row; 8 rows per 256-thread block.
// ---------------------------------------------------------------------------
__global__ __launch_bounds__(256) void logits_kernel(
    const float* __restrict__ co_dm, const float* __restrict__ addv,
    const float* __restrict__ ms_w, const float* __restrict__ ms_b,
    float* __restrict__ logits) {
  const int lane = threadIdx.x & 31;
  const int wave = threadIdx.x >> 5;
  const int row  = blockIdx.x * 8 + wave;  // row = b*kS + s
  const int b    = row >> 8;               // kS = 256

  const float* __restrict__ cr = co_dm + (size_t)row * kD;
  const float* __restrict__ ar = addv + (size_t)b * kD;

  float acc = 0.0f;
#pragma unroll 4
  for (int i = 0; i < kD / 32; ++i) {
    const int e = lane + i * 32;
    acc += tanhf(cr[e] + ar[e]) * ms_w[e];
  }
#pragma unroll
  for (int off = 16; off > 0; off >>= 1)
    acc += __shfl_down(acc, off, 32);
  if (lane == 0) logits[row] = acc + ms_b[0];
}

// ---------------------------------------------------------------------------
// Per batch b: p = softmax_s(logits[b,:]);  r[b,d] = sum_s p[s]*co[b,s,d] + rrt[b,d]
// One 256-thread block per b.
// ---------------------------------------------------------------------------
__global__ __launch_bounds__(256) void softmax_update_kernel(
    const float* __restrict__ logits, const float* __restrict__ co,
    const float* __restrict__ rrt, float* __restrict__ r) {
  __shared__ float sh[kS];
  __shared__ float p[kS];
  const int b   = blockIdx.x;
  const int tid = threadIdx.x;

  const float l = logits[b * kS + tid];
  sh[tid] = l;
  __syncthreads();
#pragma unroll
  for (int off = 128; off > 0; off >>= 1) {
    if (tid < off) sh[tid] = fmaxf(sh[tid], sh[tid + off]);
    __syncthreads();
  }
  const float mx = sh[0];
  __syncthreads();
  const float e = expf(l - mx);
  sh[tid] = e;
  __syncthreads();
#pragma unroll
  for (int off = 128; off > 0; off >>= 1) {
    if (tid < off) sh[tid] += sh[tid + off];
    __syncthreads();
  }
  const float inv = 1.0f / sh[0];
  p[tid] = e * inv;
  __syncthreads();

  const float* __restrict__ cb = co + (size_t)b * kS * kD;
#pragma unroll
  for (int rep = 0; rep < kD / 256; ++rep) {
    const int d = tid + rep * 256;
    float acc = 0.0f;
    for (int s = 0; s < kS; ++s) acc += p[s] * cb[(size_t)s * kD + d];
    r[(size_t)b * kD + d] = acc + rrt[(size_t)b * kD + d];
  }
}

__global__ void zero_kernel(float* __restrict__ p, int n) {
  const int i = blockIdx.x * blockDim.x + threadIdx.x;
  if (i < n) p[i] = 0.0f;
}

// ---------------------------------------------------------------------------
// Host-side orchestration (graph-capture safe: kernel launches only).
// ---------------------------------------------------------------------------
extern "C" void kernel_launch(void* const* d_in, const int* in_sizes, int n_in,
                              void* d_out, int out_size, void* d_ws,
                              size_t ws_size, hipStream_t stream) {
  (void)in_sizes; (void)n_in; (void)out_size; (void)ws_size;
  const float* ctx  = (const float*)d_in[0];   // [S,B,D]
  const float* qo   = (const float*)d_in[2];   // [T,B,D]
  const float* qh   = (const float*)d_in[3];   // [B,D]
  const float* img  = (const float*)d_in[4];   // [S,B,D]
  const float* fc1w = (const float*)d_in[6];
  const float* fc1b = (const float*)d_in[7];
  const float* fc2w = (const float*)d_in[8];
  const float* fc2b = (const float*)d_in[9];
  const float* dmw  = (const float*)d_in[10];
  const float* dmb  = (const float*)d_in[11];
  const float* msw  = (const float*)d_in[12];
  const float* msb  = (const float*)d_in[13];
  const float* rmw  = (const float*)d_in[14];
  const float* rmb  = (const float*)d_in[15];
  const float* qmw  = (const float*)d_in[16];
  const float* qmb  = (const float*)d_in[17];
  const float* rrw  = (const float*)d_in[18];
  const float* rrb  = (const float*)d_in[19];
  const float* rgw  = (const float*)d_in[20];
  const float* rgb  = (const float*)d_in[21];
  const float* qgw  = (const float*)d_in[22];
  const float* qgb  = (const float*)d_in[23];

  // Workspace carve-up (floats). Total ~33.8 MB.
  float* co     = (float*)d_ws;                 // [B,S,D] 4.19M
  float* co_dm  = co + (size_t)kB * kS * kD;    // [B,S,D] 4.19M
  float* addb   = co_dm + (size_t)kB * kS * kD; // [B,D]
  float* rrt    = addb + (size_t)kB * kD;       // [B,D]
  float* logitb = rrt + (size_t)kB * kD;        // [B,S]
  float* rbuf   = logitb + (size_t)kB * kS;     // [B,D]

  const int M = kS * kB;  // 8192 rows

  zero_kernel<<<(kB * kD + 255) / 256, 256, 0, stream>>>(rbuf, kB * kD);

  // co = ctx*fc1^T + img*fc2^T + fc1_b + fc2_b, remapped [S,B]->[B,S]
  big_gemm_kernel<<<M / 16, 256, 0, stream>>>(ctx, fc1w, fc1b, img, fc2w,
                                              fc2b, co, /*remap=*/1);
  // co_dm = co*dm^T + dm_b
  big_gemm_kernel<<<M / 16, 256, 0, stream>>>(co, dmw, dmb, nullptr, nullptr,
                                              nullptr, co_dm, /*remap=*/0);

  for (int t = 0; t < kT; ++t) {
    const float* qt = qo + (size_t)t * kB * kD;
    // add = r*rm^T + rm_b + q_t*qm^T + qm_b
    small_gemm_kernel<<<dim3(2, kD / 16), 32, 0, stream>>>(
        rbuf, rmw, rmb, qt, qmw, qmb, addb, /*tanh=*/0);
    // rrt = tanh(r*rr^T + rr_b)   (reads old r; runs before r is updated)
    small_gemm_kernel<<<dim3(2, kD / 16), 32, 0, stream>>>(
        rbuf, rrw, rrb, nullptr, nullptr, nullptr, rrt, /*tanh=*/1);
    logits_kernel<<<(kB * kS) / 8, 256, 0, stream>>>(co_dm, addb, msw, msb,
                                                     logitb);
    softmax_update_kernel<<<kB, 256, 0, stream>>>(logitb, co, rrt, rbuf);
  }

  // g = r*rg^T + rg_b + qh*qg^T + qg_b
  small_gemm_kernel<<<dim3(2, kD / 16), 32, 0, stream>>>(
      rbuf, rgw, rgb, qh, qgw, qgb, (float*)d_out, /*tanh=*/0);
}